// MultiHead_Self_Attention_57397942944347
// MI455X (gfx1250) — compile-verified
//
#include <hip/hip_runtime.h>
#include <hip/hip_bf16.h>
#include <math.h>

// Problem constants (from reference)
#define Bn   8
#define Sn   1024
#define D0n  128
#define Dn   256
#define Hn   8
#define HDn  32
#define INV_SCALE 0.17677669529663687f   // 1/sqrt(32)

#define QKV_ELEMS (Bn*Hn*Sn*HDn)   // 2,097,152 floats per tensor
#define OUTN      (Bn*Sn*D0n)      // output floats; attn follows in d_out

typedef float v2f __attribute__((ext_vector_type(2)));
typedef float v4f __attribute__((ext_vector_type(4)));
typedef float v8f __attribute__((ext_vector_type(8)));

static __device__ __forceinline__ v8f zero8() {
  v8f z = {0.f,0.f,0.f,0.f,0.f,0.f,0.f,0.f};
  return z;
}

// D = A(16x4,f32) * B(4x16,f32) + C(16x16,f32)  — CDNA5 V_WMMA_F32_16X16X4_F32
static __device__ __forceinline__ v8f wmma4(v2f a, v2f b, v8f c) {
  return __builtin_amdgcn_wmma_f32_16x16x4_f32(false, a, false, b, (short)0, c,
                                               false, false);
}

// ---------------------------------------------------------------------------
// Kernel 1: fused Q/K/V projection.  Y = X @ W^T with X=(8192,128), W=(256,128).
// One wave per 16x16 output tile; 3 accumulators share A-tile loads.
// Also writes per-row Q partial sums (per half-head) for the key mask.
// ---------------------------------------------------------------------------
__global__ __launch_bounds__(128) void k_qkv(
    const float* __restrict__ x,
    const float* __restrict__ Wq, const float* __restrict__ Wk,
    const float* __restrict__ Wv,
    float* __restrict__ Qw, float* __restrict__ Kw, float* __restrict__ Vw,
    float* __restrict__ qsum2)
{
  const int wid  = blockIdx.x * 4 + (threadIdx.x >> 5);  // 0..8191
  const int lane = threadIdx.x & 31;
  const int l16  = lane & 15;
  const int half = lane >> 4;
  const int mt = wid >> 4;        // 0..511  (16-row tile of 8192)
  const int nt = wid & 15;        // 0..15   (16-col tile of 256)
  const int m0 = mt * 16, n0 = nt * 16;

  const float* arow = x  + (size_t)(m0 + l16) * D0n + 2 * half;
  const float* bqr  = Wq + (size_t)(n0 + l16) * D0n + 2 * half;
  const float* bkr  = Wk + (size_t)(n0 + l16) * D0n + 2 * half;
  const float* bvr  = Wv + (size_t)(n0 + l16) * D0n + 2 * half;

  v8f cq = zero8(), ck = zero8(), cv = zero8();
  #pragma unroll 4
  for (int kc = 0; kc < 32; ++kc) {
    const int k = kc * 4;
    v2f a  = *(const v2f*)(arow + k);
    v2f b0 = *(const v2f*)(bqr + k);
    v2f b1 = *(const v2f*)(bkr + k);
    v2f b2 = *(const v2f*)(bvr + k);
    cq = wmma4(a, b0, cq);
    ck = wmma4(a, b1, ck);
    cv = wmma4(a, b2, cv);
  }

  const int h   = n0 >> 5;      // head of this 16-col tile (constant in tile)
  const int sel = nt & 1;       // which half of the head
  const int d   = (n0 + l16) & 31;

  #pragma unroll
  for (int i = 0; i < 8; ++i) {
    const int row = m0 + i + 8 * half;     // C layout: vgpr i, lane halves
    const int bb = row >> 10, ss = row & 1023;
    const size_t idx = (((size_t)bb * Hn + h) * Sn + ss) * HDn + d;
    Qw[idx] = cq[i]; Kw[idx] = ck[i]; Vw[idx] = cv[i];
  }

  // Partial row sums of Q over this tile's 16 columns (16-lane reduction).
  #pragma unroll
  for (int i = 0; i < 8; ++i) {
    float v = cq[i];
    v += __shfl_xor(v, 1, 32);
    v += __shfl_xor(v, 2, 32);
    v += __shfl_xor(v, 4, 32);
    v += __shfl_xor(v, 8, 32);
    if (l16 == 0) {
      const int row = m0 + i + 8 * half;
      const int bb = row >> 10, ss = row & 1023;
      qsum2[(((size_t)bb * Hn + h) * Sn + ss) * 2 + sel] = v;
    }
  }
}

// ---------------------------------------------------------------------------
// Kernel 2: attention. One 128-thread workgroup (4 waves) per (b,h,16-row q
// tile); the 16x1024 score strip is shared in LDS (~75 KB/WG -> 4 WGs/WGP).
// ---------------------------------------------------------------------------
#define SROW 1028                        // row stride in floats (bank spread)
#define OFF_MASKB (16 * SROW)            // 1024 floats: additive mask bias
#define OFF_PMAX  (OFF_MASKB + 1024)     // 128 floats: per-(row,seg) max
#define OFF_PSUM  (OFF_PMAX + 128)       // 128 floats: per-(row,seg) expsum
#define OFF_RMAX  (OFF_PSUM + 128)       // 16 floats
#define OFF_RSUM  (OFF_RMAX + 16)        // 16 floats
#define OFF_PACC  (OFF_RSUM + 16)        // 4*256 floats: attn@V partials
#define LDS_FLOATS (OFF_PACC + 1024)     // 18784 floats = 75136 bytes

__global__ __launch_bounds__(128) void k_attn(
    const float* __restrict__ U,
    const float* __restrict__ Qw, const float* __restrict__ Kw,
    const float* __restrict__ Vw,
    const float* __restrict__ qsum2,
    float* __restrict__ attn,            // d_out + OUTN, (B,H,S,S)
    float* __restrict__ ctx)             // (B,H,S,HD)
{
  extern __shared__ float smem[];
  float* slab   = smem;
  float* maskb  = smem + OFF_MASKB;
  float* pmax   = smem + OFF_PMAX;
  float* psum   = smem + OFF_PSUM;
  float* rowmax = smem + OFF_RMAX;
  float* rowsum = smem + OFF_RSUM;
  float* pacc   = smem + OFF_PACC;

  const int t     = threadIdx.x;
  const int wslot = t >> 5;
  const int lane  = t & 31;
  const int l16   = lane & 15;
  const int half  = lane >> 4;
  const int wid = blockIdx.x;            // 0..4095
  const int bh  = wid >> 6;              // b*H + h
  const int m0  = (wid & 63) * 16;

  const float* Ub = U + ((size_t)bh * Sn + m0) * Sn;

  // ---- Phase 0: mask bias (0 / -inf) per key column; prefetch U strip ----
  for (int j = t; j < Sn; j += 128) {
    const float qs = qsum2[((size_t)bh * Sn + j) * 2] +
                     qsum2[((size_t)bh * Sn + j) * 2 + 1];
    maskb[j] = (qs == 0.0f) ? -INFINITY : 0.0f;
  }
  for (int line = t; line < 512; line += 128)          // 512 x 128B = 64 KB
    __builtin_prefetch(Ub + (size_t)line * 32, 0, 0);  // global_prefetch_b8

  // ---- Phase 1: raw Q K^T tiles -> slab (waves split the 64 key tiles) ----
  const float* Qb = Qw + ((size_t)bh * Sn + m0) * HDn;
  v2f a[8];
  #pragma unroll
  for (int kc = 0; kc < 8; ++kc)
    a[kc] = *(const v2f*)(Qb + l16 * HDn + kc * 4 + 2 * half);

  for (int ii = 0; ii < 16; ++ii) {
    const int jt = wslot + ii * 4;
    const int j0 = jt * 16;
    const float* Kb = Kw + ((size_t)bh * Sn + j0 + l16) * HDn + 2 * half;
    v8f c = zero8();
    #pragma unroll
    for (int kc = 0; kc < 8; ++kc) {
      v2f b = *(const v2f*)(Kb + kc * 4);
      c = wmma4(a[kc], b, c);
    }
    #pragma unroll
    for (int i = 0; i < 8; ++i)
      slab[(i + 8 * half) * SROW + j0 + l16] = c[i];
  }
  __syncthreads();

  // ---- Phase 1b: s = (qk + U)*scale + maskbias  (coalesced b128, NT) ----
  for (int idx = t; idx < 4096; idx += 128) {
    const int r  = idx >> 8;
    const int jc = (idx & 255) * 4;
    v4f u  = __builtin_nontemporal_load((const v4f*)(Ub + (size_t)r * Sn + jc));
    v4f s  = *(const v4f*)(slab + r * SROW + jc);
    v4f mb = *(const v4f*)(maskb + jc);
    s = (s + u) * INV_SCALE + mb;
    *(v4f*)(slab + r * SROW + jc) = s;
  }
  __syncthreads();

  // ---- Phase 2: row max, then exp-in-place + row sum (b128 LDS, 4 accs) ----
  {
    const int r = t & 15, seg = t >> 4;                // 8 segs x 128 elems
    const float* row = slab + r * SROW + seg * 128;
    v4f m4 = { -INFINITY, -INFINITY, -INFINITY, -INFINITY };
    #pragma unroll 4
    for (int j = 0; j < 32; ++j) {
      v4f v = *(const v4f*)(row + j * 4);
      m4.x = fmaxf(m4.x, v.x); m4.y = fmaxf(m4.y, v.y);
      m4.z = fmaxf(m4.z, v.z); m4.w = fmaxf(m4.w, v.w);
    }
    pmax[r * 8 + seg] = fmaxf(fmaxf(m4.x, m4.y), fmaxf(m4.z, m4.w));
  }
  __syncthreads();
  if (t < 16) {
    float mx = pmax[t * 8];
    #pragma unroll
    for (int s = 1; s < 8; ++s) mx = fmaxf(mx, pmax[t * 8 + s]);
    rowmax[t] = mx;
  }
  __syncthreads();
  {
    const int r = t & 15, seg = t >> 4;
    const float mx = rowmax[r];
    float* row = slab + r * SROW + seg * 128;
    v4f s4 = { 0.f, 0.f, 0.f, 0.f };
    #pragma unroll 4
    for (int j = 0; j < 32; ++j) {
      v4f v = *(const v4f*)(row + j * 4);
      v.x = __expf(v.x - mx); v.y = __expf(v.y - mx);
      v.z = __expf(v.z - mx); v.w = __expf(v.w - mx);
      *(v4f*)(row + j * 4) = v;
      s4 = s4 + v;
    }
    psum[r * 8 + seg] = (s4.x + s4.y) + (s4.z + s4.w);
  }
  __syncthreads();
  if (t < 16) {
    float sm = psum[t * 8];
    #pragma unroll
    for (int s = 1; s < 8; ++s) sm += psum[t * 8 + s];
    rowsum[t] = sm;
  }
  __syncthreads();

  // ---- Phase 3: normalize; write attn (NT, b128) and back to slab ----
  float* attnB = attn + ((size_t)bh * Sn + m0) * Sn;
  for (int idx = t; idx < 4096; idx += 128) {
    const int r  = idx >> 8;
    const int jc = (idx & 255) * 4;
    const float is = 1.0f / rowsum[r];
    v4f v = *(const v4f*)(slab + r * SROW + jc);
    v = v * is;
    *(v4f*)(slab + r * SROW + jc) = v;
    __builtin_nontemporal_store(v, (v4f*)(attnB + (size_t)r * Sn + jc));
  }
  __syncthreads();

  // ---- Phase 4: ctx = attn @ V; waves split (d-tile x key-half) ----
  v8f o = zero8();
  const int ntile = wslot & 1;           // d cols 0..15 or 16..31
  const int dbase = ntile * 16;
  const int jtb   = (wslot >> 1) * 32;   // key tiles [jtb, jtb+32)
  const float* Vb = Vw + (size_t)bh * Sn * HDn;
  for (int jt = jtb; jt < jtb + 32; ++jt) {
    #pragma unroll
    for (int c4 = 0; c4 < 4; ++c4) {
      const int kk = jt * 16 + c4 * 4;
      v2f av = *(const v2f*)(slab + l16 * SROW + kk + 2 * half);
      const float* vr = Vb + (size_t)(kk + 2 * half) * HDn + dbase;
      v2f b = { vr[l16], vr[HDn + l16] };
      o = wmma4(av, b, o);
    }
  }
  #pragma unroll
  for (int i = 0; i < 8; ++i)
    pacc[wslot * 256 + (i + 8 * half) * 16 + l16] = o[i];
  __syncthreads();

  float* ctxB = ctx + ((size_t)bh * Sn + m0) * HDn;
  for (int idx = t; idx < 512; idx += 128) {
    const int r = idx >> 5, d = idx & 31;
    const int nt2 = d >> 4, c = d & 15;
    const float v = pacc[nt2 * 256 + r * 16 + c] +
                    pacc[(nt2 + 2) * 256 + r * 16 + c];
    ctxB[(size_t)r * HDn + d] = v;
  }
}

// ---------------------------------------------------------------------------
// Kernel 3: output projection: out(8192,128) = ctx(8192,256) @ Wo^T
// ---------------------------------------------------------------------------
__global__ __launch_bounds__(128) void k_oproj(
    const float* __restrict__ ctx, const float* __restrict__ Wo,
    float* __restrict__ out)
{
  const int wid  = blockIdx.x * 4 + (threadIdx.x >> 5);  // 0..4095
  const int lane = threadIdx.x & 31;
  const int l16  = lane & 15;
  const int half = lane >> 4;
  const int mt = wid >> 3;     // 0..511
  const int nt = wid & 7;      // 0..7
  const int m0 = mt * 16, n0 = nt * 16;
  const int row = m0 + l16;
  const int bb = row >> 10, ss = row & 1023;

  const float* brow = Wo + (size_t)(n0 + l16) * Dn + 2 * half;
  v8f c = zero8();
  #pragma unroll 4
  for (int kc = 0; kc < 64; ++kc) {
    const int k = kc * 4;
    const int h = k >> 5, d = k & 31;     // ctx is (B,H,S,HD)
    v2f a = *(const v2f*)(ctx + (((size_t)bb * Hn + h) * Sn + ss) * HDn + d + 2 * half);
    v2f b = *(const v2f*)(brow + k);
    c = wmma4(a, b, c);
  }
  #pragma unroll
  for (int i = 0; i < 8; ++i) {
    const int r = m0 + i + 8 * half;
    out[(size_t)r * D0n + n0 + l16] = c[i];
  }
}

// ---------------------------------------------------------------------------
extern "C" void kernel_launch(void* const* d_in, const int* in_sizes, int n_in,
                              void* d_out, int out_size, void* d_ws, size_t ws_size,
                              hipStream_t stream)
{
  const float* query = (const float*)d_in[0];
  const float* U     = (const float*)d_in[1];
  const float* Wq    = (const float*)d_in[2];
  const float* Wk    = (const float*)d_in[3];
  const float* Wv    = (const float*)d_in[4];
  const float* Wo    = (const float*)d_in[5];

  float* out  = (float*)d_out;          // (B,S,D0)
  float* attn = out + OUTN;             // (B,H,S,S)

  float* w      = (float*)d_ws;
  float* Qw     = w;
  float* Kw     = Qw + QKV_ELEMS;
  float* Vw     = Kw + QKV_ELEMS;
  float* ctx    = Vw + QKV_ELEMS;
  float* qsum2  = ctx + QKV_ELEMS;      // B*H*S*2 floats

  // 1) QKV projection: 8192 waves, 4 waves per block
  k_qkv<<<2048, 128, 0, stream>>>(query, Wq, Wk, Wv, Qw, Kw, Vw, qsum2);

  // 2) Attention: one WG per (b,h,qtile); ~75 KB dynamic LDS -> 4 WGs/WGP
  const size_t shbytes = (size_t)LDS_FLOATS * sizeof(float);
  k_attn<<<4096, 128, shbytes, stream>>>(U, Qw, Kw, Vw, qsum2, attn, ctx);

  // 3) Output projection: 4096 waves
  k_oproj<<<1024, 128, 0, stream>>>(ctx, Wo, out);
}